// Congestion_prediction_16561393894030
// MI455X (gfx1250) — compile-verified
//
#include <hip/hip_runtime.h>
#include <hip/hip_bf16.h>

// RUDY congestion maps for gfx1250 (MI455X).
// Grid: NBX x NBY bins, BX = BY = 1.0 exactly for this problem.
#define NBX 168
#define NBY 480
#define XLC 0.0f
#define YLC 0.0f
#define BXC 1.0f
#define BYC 1.0f
#define UNIT_HCAP 1.5f
#define UNIT_VCAP 1.4f
#define NMAPS (NBX * NBY)

typedef __attribute__((ext_vector_type(2))) float v2f;
typedef __attribute__((ext_vector_type(8))) float v8f;

struct NetRec {
    float xmin, xmax, ymin, ymax;
    float dh, dv;
    int il, jl;
};

__global__ void zero_maps_kernel(float* __restrict__ maps, int n) {
    int i = blockIdx.x * blockDim.x + threadIdx.x;
    if (i < n) maps[i] = 0.0f;
}

// One thread per net: bbox over its pins + density weights + window origin.
__global__ void net_bbox_kernel(const float2* __restrict__ pin_pos,
                                const float* __restrict__ net_weights,
                                const int* __restrict__ netpin_start,
                                const int* __restrict__ flat_netpin,
                                NetRec* __restrict__ recs, int N) {
    int n = blockIdx.x * blockDim.x + threadIdx.x;
    if (n >= N) return;
    int s = netpin_start[n];
    int e = netpin_start[n + 1];
    NetRec r;
    if (s >= e) {  // invalid net: dh=dv=0 -> zero contribution downstream
        r.xmin = 1.0f; r.xmax = 0.0f; r.ymin = 1.0f; r.ymax = 0.0f;
        r.dh = 0.0f; r.dv = 0.0f; r.il = 0; r.jl = 0;
        recs[n] = r;
        return;
    }
    float xmin = 3.4e38f, xmax = -3.4e38f, ymin = 3.4e38f, ymax = -3.4e38f;
    for (int k = s; k < e; ++k) {
        int p = flat_netpin[k];
        float2 xy = pin_pos[p];
        xmin = fminf(xmin, xy.x);
        xmax = fmaxf(xmax, xy.x);
        ymin = fminf(ymin, xy.y);
        ymax = fmaxf(ymax, xy.y);
    }
    float w = net_weights[n];
    r.xmin = xmin; r.xmax = xmax; r.ymin = ymin; r.ymax = ymax;
    r.dh = w / fmaxf(ymax - ymin, BYC);
    r.dv = w / fmaxf(xmax - xmin, BXC);
    r.il = min(max((int)floorf((xmin - XLC) / BXC), 0), NBX - 1);
    r.jl = min(max((int)floorf((ymin - YLC) / BYC), 0), NBY - 1);
    recs[n] = r;
}

__device__ __forceinline__ float ovx_of(float xmin, float xmax, int il, int a) {
    int bi = il + a;
    float lox = XLC + (float)bi * BXC;
    float v = fminf(xmax, lox + BXC) - fmaxf(xmin, lox);
    v = fminf(fmaxf(v, 0.0f), BXC);
    return (bi < NBX) ? v : 0.0f;
}

__device__ __forceinline__ float ovy_of(float ymin, float ymax, int jl, int b) {
    int bj = jl + b;
    float loy = YLC + (float)bj * BYC;
    float v = fminf(ymax, loy + BYC) - fmaxf(ymin, loy);
    v = fminf(fmaxf(v, 0.0f), BYC);
    return (bj < NBY) ? v : 0.0f;
}

// One wave32 per pair of nets. Block-diagonal rank-1 outer products via
// V_WMMA_F32_16X16X4_F32:
//   A[m][0] = dh0*ovx0[m]         (m<8)     A[m][1] = dh1*ovx1[m-8] (m>=8)
//   B[0][n] = ovy0[n]             (n<8)     B[1][n] = ovy1[n-8]     (n>=8)
// -> C(M<8,N<8)  = net0 H-area;   C(M>=8,N>=8) = net1 H-area; cross blocks 0.
// Second WMMA with dv-scaled A produces V-areas. C/D layout maps VGPR r,
// lane -> (M = r + 8*(lane>=16), N = lane&15), so each active lane owns one
// bin column and scatters 8 rows via f32 global atomics (L2-resident maps).
__global__ void rudy_scatter_wmma_kernel(const NetRec* __restrict__ recs,
                                         float* __restrict__ H,
                                         float* __restrict__ V,
                                         int N, int numPairs) {
    int gtid = blockIdx.x * blockDim.x + threadIdx.x;
    int wave = gtid >> 5;          // wave-uniform pair index
    int lane = threadIdx.x & 31;
    if (wave >= numPairs) return;  // uniform across the wave -> EXEC full below

    int n0 = 2 * wave;
    int n1 = n0 + 1;
    NetRec r0 = recs[n0];
    NetRec r1;
    if (n1 < N) {
        r1 = recs[n1];
    } else {
        r1.xmin = 1.0f; r1.xmax = 0.0f; r1.ymin = 1.0f; r1.ymax = 0.0f;
        r1.dh = 0.0f; r1.dv = 0.0f; r1.il = 0; r1.jl = 0;
    }

    // A (16x4 f32): VGPR0 = K0 (lanes 0-15) / K2 (lanes 16-31); VGPR1 = K1/K3.
    // K2,K3 kept zero. B (4x16) mirrors this layout.
    v2f aH = {0.0f, 0.0f};
    v2f aV = {0.0f, 0.0f};
    v2f bm = {0.0f, 0.0f};
    if (lane < 16) {
        int m = lane;
        if (m < 8) {
            float ox = ovx_of(r0.xmin, r0.xmax, r0.il, m);
            aH.x = r0.dh * ox;
            aV.x = r0.dv * ox;
            bm.x = ovy_of(r0.ymin, r0.ymax, r0.jl, m);   // B[0][n], n = lane
        } else {
            float ox = ovx_of(r1.xmin, r1.xmax, r1.il, m - 8);
            aH.y = r1.dh * ox;
            aV.y = r1.dv * ox;
            bm.y = ovy_of(r1.ymin, r1.ymax, r1.jl, m - 8);  // B[1][n]
        }
    }

    v8f cH = {};
    v8f cV = {};
    cH = __builtin_amdgcn_wmma_f32_16x16x4_f32(false, aH, false, bm,
                                               (short)0, cH, false, false);
    cV = __builtin_amdgcn_wmma_f32_16x16x4_f32(false, aV, false, bm,
                                               (short)0, cV, false, false);

    int half = lane >> 4;
    int ncol = lane & 15;
    if (half == 0 && ncol < 8) {           // net0 block: M=r, N=ncol
        int bj = min(r0.jl + ncol, NBY - 1);
#pragma unroll
        for (int r = 0; r < 8; ++r) {
            int bi = min(r0.il + r, NBX - 1);
            int idx = bi * NBY + bj;
            float hv = cH[r];
            float vv = cV[r];
            if (hv != 0.0f) atomicAdd(&H[idx], hv);
            if (vv != 0.0f) atomicAdd(&V[idx], vv);
        }
    } else if (half == 1 && ncol >= 8) {   // net1 block: M=r+8, N=ncol
        int b = ncol - 8;
        int bj = min(r1.jl + b, NBY - 1);
#pragma unroll
        for (int r = 0; r < 8; ++r) {
            int bi = min(r1.il + r, NBX - 1);
            int idx = bi * NBY + bj;
            float hv = cH[r];
            float vv = cV[r];
            if (hv != 0.0f) atomicAdd(&H[idx], hv);
            if (vv != 0.0f) atomicAdd(&V[idx], vv);
        }
    }
}

// flat_netpin is a permutation of all pins, so pin density = histogram over
// pin_pos directly.
__global__ void pin_density_kernel(const float2* __restrict__ pin_pos,
                                   float* __restrict__ Pd, int P) {
    int p = blockIdx.x * blockDim.x + threadIdx.x;
    if (p >= P) return;
    float2 xy = pin_pos[p];
    int pi = min(max((int)floorf((xy.x - XLC) / BXC), 0), NBX - 1);
    int pj = min(max((int)floorf((xy.y - YLC) / BYC), 0), NBY - 1);
    atomicAdd(&Pd[pi * NBY + pj], 1.0f);
}

__global__ void finalize_kernel(const float* __restrict__ H,
                                const float* __restrict__ V,
                                const float* __restrict__ Pd,
                                float* __restrict__ out) {
    int i = blockIdx.x * blockDim.x + threadIdx.x;
    if (i >= NMAPS) return;
    float pd = Pd[i] * (1.0f / 250.0f);
    float h2 = H[i] * (1.0f / 512.0f) + 0.5f * pd;
    float v2 = V[i] * (1.0f / 512.0f) + 0.5f * pd;
    float hr = h2 * (512.0f / UNIT_HCAP);
    float vr = v2 * (512.0f / UNIT_VCAP);
    out[i] = fmaxf(fabsf(hr), fabsf(vr));   // route
    out[NMAPS + i] = hr;                    // h_res
    out[2 * NMAPS + i] = vr;                // v_res
}

extern "C" void kernel_launch(void* const* d_in, const int* in_sizes, int n_in,
                              void* d_out, int out_size, void* d_ws, size_t ws_size,
                              hipStream_t stream) {
    const float2* pin_pos     = (const float2*)d_in[0];  // (P,2) f32
    const float*  net_weights = (const float*)d_in[1];   // (N,)  f32
    const int*    netpin_start = (const int*)d_in[2];    // (N+1,) i32
    const int*    flat_netpin  = (const int*)d_in[3];    // (P,)  i32
    // d_in[4] = pin_directs: unused by the reference.

    int N = in_sizes[2] - 1;
    int P = in_sizes[3];

    // Workspace layout: H | V | Pd | per-net records (16B aligned).
    float* H  = (float*)d_ws;
    float* V  = H + NMAPS;
    float* Pd = V + NMAPS;
    NetRec* recs = (NetRec*)(Pd + NMAPS);

    zero_maps_kernel<<<(3 * NMAPS + 255) / 256, 256, 0, stream>>>(H, 3 * NMAPS);

    net_bbox_kernel<<<(N + 255) / 256, 256, 0, stream>>>(
        pin_pos, net_weights, netpin_start, flat_netpin, recs, N);

    int numPairs = (N + 1) / 2;
    long long threads = (long long)numPairs * 32;
    rudy_scatter_wmma_kernel<<<(int)((threads + 255) / 256), 256, 0, stream>>>(
        recs, H, V, N, numPairs);

    pin_density_kernel<<<(P + 255) / 256, 256, 0, stream>>>(pin_pos, Pd, P);

    finalize_kernel<<<(NMAPS + 255) / 256, 256, 0, stream>>>(
        H, V, Pd, (float*)d_out);
}